// LightGCN_17334488006881
// MI455X (gfx1250) — compile-verified
//
#include <hip/hip_runtime.h>
#include <hip/hip_bf16.h>

#define TILE 256
#define EMBED_DIM 64
#define SCATTER_BLOCKS 2048

// ---------------------------------------------------------------------------
// init: cur = embedding ; acc(d_out) = embedding     (float4 vectorized)
// ---------------------------------------------------------------------------
__global__ void lgcn_init(const float4* __restrict__ emb,
                          float4* __restrict__ cur,
                          float4* __restrict__ acc, int n4) {
  int i = blockIdx.x * blockDim.x + threadIdx.x;
  if (i < n4) {
    float4 v = emb[i];
    cur[i] = v;
    acc[i] = v;
  }
}

// ---------------------------------------------------------------------------
// zero: p = 0
// ---------------------------------------------------------------------------
__global__ void lgcn_zero(float4* __restrict__ p, int n4) {
  int i = blockIdx.x * blockDim.x + threadIdx.x;
  if (i < n4) p[i] = make_float4(0.f, 0.f, 0.f, 0.f);
}

// ---------------------------------------------------------------------------
// accum: acc += E ; on last layer acc = (acc + E) * 0.25
// ---------------------------------------------------------------------------
__global__ void lgcn_accum(float4* __restrict__ acc,
                           const float4* __restrict__ E,
                           int n4, int last) {
  int i = blockIdx.x * blockDim.x + threadIdx.x;
  if (i < n4) {
    float4 a = acc[i];
    float4 b = E[i];
    a.x += b.x; a.y += b.y; a.z += b.z; a.w += b.w;
    if (last) { a.x *= 0.25f; a.y *= 0.25f; a.z *= 0.25f; a.w *= 0.25f; }
    acc[i] = a;
  }
}

// ---------------------------------------------------------------------------
// scatter: E_next[dst] += w * E[src]
//
// Grid-stride over 256-edge tiles with DOUBLE-BUFFERED async staging:
//   - thread t async-loads (src,dst,w)[tile*256+t] into LDS buf (ASYNCcnt)
//   - next tile's 3 asyncs are issued before waiting; async loads retire
//     in order, so `s_wait_asynccnt 3` guarantees the current buffer's
//     3 oldest requests have landed while the prefetch stays in flight.
//   - every wave always issues exactly 3 asyncs per stage (addresses are
//     clamped, never predicated) so the counter accounting is exact.
//   - compute: half-wave (16 lanes) per edge, lane l covers dims [4l,4l+3]
//     -> one coalesced 256B global_load_b128 gather per edge,
//        4 native global_atomic_add_f32 (no-return) per lane.
// ---------------------------------------------------------------------------
__global__ __launch_bounds__(TILE) void lgcn_scatter(
    const float* __restrict__ E, float* __restrict__ Enext,
    const int* __restrict__ src, const int* __restrict__ dst,
    const float* __restrict__ w, int n_edges, int n_tiles) {
  __shared__ int   s_src[2][TILE];
  __shared__ int   s_dst[2][TILE];
  __shared__ float s_w[2][TILE];

  const int t = threadIdx.x;
  const int G = gridDim.x;

  auto stage = [&](int tile, int buf) {
    int e = tile * TILE + t;
    unsigned ee  = (e < n_edges) ? (unsigned)e : (unsigned)(n_edges - 1);
    unsigned off = ee * 4u;                                  // byte offset
    unsigned a0 = (unsigned)(uintptr_t)(&s_src[buf][t]);     // LDS byte addrs
    unsigned a1 = (unsigned)(uintptr_t)(&s_dst[buf][t]);
    unsigned a2 = (unsigned)(uintptr_t)(&s_w[buf][t]);
    // GVS mode: saddr = 64-bit SGPR base, vaddr = 32-bit per-lane byte offset
    asm volatile("global_load_async_to_lds_b32 %0, %1, %2"
                 :: "v"(a0), "v"(off), "s"(src) : "memory");
    asm volatile("global_load_async_to_lds_b32 %0, %1, %2"
                 :: "v"(a1), "v"(off), "s"(dst) : "memory");
    asm volatile("global_load_async_to_lds_b32 %0, %1, %2"
                 :: "v"(a2), "v"(off), "s"(w) : "memory");
  };

  int tile = blockIdx.x;
  if (tile < n_tiles) stage(tile, 0);

  const int g = t >> 4;          // edge slot group (0..15)
  const int d = (t & 15) << 2;   // dim offset (0,4,...,60)

  int buf = 0;
  for (; tile < n_tiles; tile += G, buf ^= 1) {
    const bool pf = (tile + G) < n_tiles;
    if (pf) stage(tile + G, buf ^ 1);          // prefetch into other buffer

    if (pf) asm volatile("s_wait_asynccnt 0x3" ::: "memory");
    else    asm volatile("s_wait_asynccnt 0x0" ::: "memory");
    __syncthreads();                            // all waves' cur buf staged

    const int base = tile * TILE;
    const int rem  = n_edges - base;
    const int lim  = (rem < TILE) ? rem : TILE;
#pragma unroll 2
    for (int i = g; i < lim; i += 16) {
      const int   s  = s_src[buf][i];           // LDS same-addr broadcast
      const int   dd = s_dst[buf][i];
      const float wt = s_w[buf][i];
      const float4 v =
          *reinterpret_cast<const float4*>(E + (size_t)s * EMBED_DIM + d);
      float* o = Enext + (size_t)dd * EMBED_DIM + d;
      unsafeAtomicAdd(o + 0, v.x * wt);         // global_atomic_add_f32
      unsafeAtomicAdd(o + 1, v.y * wt);
      unsafeAtomicAdd(o + 2, v.z * wt);
      unsafeAtomicAdd(o + 3, v.w * wt);
    }
    __syncthreads();                            // cur buf free for re-staging
  }
}

// ---------------------------------------------------------------------------
// launch: init -> 3x [zero, scatter, accum(last scales by 1/4)]
// d_ws: ping-pong layer buffers (2 * 140000*64 floats = 71.7 MB)
// d_out: running accumulator -> final mean, rows 0..N-1 = user||item concat
// ---------------------------------------------------------------------------
extern "C" void kernel_launch(void* const* d_in, const int* in_sizes, int n_in,
                              void* d_out, int out_size, void* d_ws, size_t ws_size,
                              hipStream_t stream) {
  const float* emb = (const float*)d_in[0];   // (140000, 64) f32
  const float* wgt = (const float*)d_in[1];   // (4M,) f32
  const int*   ei  = (const int*)d_in[2];     // (2, 4M) int32

  const int n_emb   = in_sizes[0];            // 8,960,000 floats
  const int n_edges = in_sizes[1];            // 4,000,000
  const int* src = ei;
  const int* dst = ei + n_edges;

  const int n4 = n_emb / 4;
  float* buf0 = (float*)d_ws;
  float* buf1 = buf0 + n_emb;
  float* acc  = (float*)d_out;

  const int eb = 256;
  const int gb = (n4 + eb - 1) / eb;
  const int n_tiles = (n_edges + TILE - 1) / TILE;
  const int sb = (n_tiles < SCATTER_BLOCKS) ? n_tiles : SCATTER_BLOCKS;

  lgcn_init<<<gb, eb, 0, stream>>>((const float4*)emb, (float4*)buf0,
                                   (float4*)acc, n4);

  for (int k = 0; k < 3; ++k) {
    float* cur = (k & 1) ? buf1 : buf0;
    float* nxt = (k & 1) ? buf0 : buf1;
    lgcn_zero<<<gb, eb, 0, stream>>>((float4*)nxt, n4);
    lgcn_scatter<<<sb, TILE, 0, stream>>>(cur, nxt, src, dst, wgt,
                                          n_edges, n_tiles);
    lgcn_accum<<<gb, eb, 0, stream>>>((float4*)acc, (const float4*)nxt, n4,
                                      (k == 2) ? 1 : 0);
  }
}